// GATv2IDNet_3169685865386
// MI455X (gfx1250) — compile-verified
//
#include <hip/hip_runtime.h>
#include <hip/hip_bf16.h>

typedef float v8f __attribute__((ext_vector_type(8)));
typedef float v2f __attribute__((ext_vector_type(2)));

#define HEADS 8
#define HD    32
#define CDIM  256
#define KMAX  256

// ---------------- helpers ----------------
__device__ __forceinline__ unsigned f2key(float f) {
    unsigned u = __float_as_uint(f);
    return (u & 0x80000000u) ? ~u : (u | 0x80000000u);
}
__device__ __forceinline__ float key2f(unsigned k) {
    unsigned u = (k & 0x80000000u) ? (k & 0x7FFFFFFFu) : ~k;
    return __uint_as_float(u);
}

// ---------------- input projection: h0 = x @ W_in + b_in  (N x 18 @ 18 x 32)
__global__ __launch_bounds__(256) void k_in_proj(const float* __restrict__ x,
                                                 const float* __restrict__ W,
                                                 const float* __restrict__ b,
                                                 float* __restrict__ h0, int N) {
    int i = blockIdx.x * blockDim.x + threadIdx.x;
    if (i >= N * 32) return;
    int n = i >> 5, c = i & 31;
    float s = b[c];
    const float* xp = x + n * 18;
#pragma unroll
    for (int k = 0; k < 18; ++k) s += xp[k] * W[k * 32 + c];
    h0[i] = s;
}

// ---------------- fp32 WMMA GEMM: C[MxNC] = A[MxK] @ B[KxNC] (+bias) ----------
// One wave32 computes one 16x16 tile with V_WMMA_F32_16X16X4_F32.
// B tile (K x 16) is staged once per block into LDS, transposed so each lane's
// fragment is one aligned ds_load_b64. K must be a multiple of 4 and <= KMAX.
// blockDim = 256 (8 waves along M); grid = (ceil(M/128), ceil(NC/16)).
__global__ __launch_bounds__(256) void k_gemm_f32_wmma(const float* __restrict__ A,
                                                       const float* __restrict__ B,
                                                       const float* __restrict__ bias,
                                                       float* __restrict__ C,
                                                       int M, int K, int NC) {
    __shared__ float Bs[KMAX * 16];          // Bs[c*K + k], c = 0..15

    const int tid  = threadIdx.x;
    const int lane = tid & 31;
    const int wave = tid >> 5;
    const int tn   = blockIdx.y;
    const int tm   = blockIdx.x * 8 + wave;

    // ---- stage B tile (coalesced: consecutive tid -> consecutive columns) ----
    for (int idx = tid; idx < K * 16; idx += 256) {
        int k = idx >> 4, c = idx & 15;
        int col = tn * 16 + c;
        Bs[c * K + k] = (col < NC) ? B[(long)k * NC + col] : 0.f;
    }
    __syncthreads();

    if (tm * 16 >= M) return;                // wave-uniform; after the only barrier

    const int r    = lane & 15;
    const int half = lane >> 4;              // lanes 0-15: K=(k,k+1); 16-31: K=(k+2,k+3)
    const float* arow = A + (long)(tm * 16 + r) * K;
    const float* brow = Bs + r * K;

    v8f acc = {0.f, 0.f, 0.f, 0.f, 0.f, 0.f, 0.f, 0.f};
#pragma unroll 8
    for (int k = 0; k < K; k += 4) {
        int ka = k + 2 * half;
        v2f a = *(const v2f*)(arow + ka);    // global_load_b64
        v2f b = *(const v2f*)(brow + ka);    // ds_load_b64
        acc = __builtin_amdgcn_wmma_f32_16x16x4_f32(false, a, false, b,
                                                    (short)0, acc, false, false);
    }

    // D layout: VGPR v -> M = half*8 + v, N = lane&15
    const int nOut  = tn * 16 + r;
    const int mBase = tm * 16 + half * 8;
    if (nOut < NC) {
        float bv = bias ? bias[nOut] : 0.f;
#pragma unroll
        for (int v = 0; v < 8; ++v)
            C[(long)(mBase + v) * NC + nOut] = acc[v] + bv;
    }
}

// ---------------- edge pass 1: raw scores + segment max (uint-key atomicMax) --
__global__ __launch_bounds__(256) void k_edge_score(const float* __restrict__ xl,
                                                    const float* __restrict__ xr,
                                                    const float* __restrict__ ea,
                                                    const int* __restrict__ src,
                                                    const int* __restrict__ dst,
                                                    const float* __restrict__ We,
                                                    const float* __restrict__ att,
                                                    float* __restrict__ score,
                                                    unsigned* __restrict__ smaxKey,
                                                    int E) {
    int i = blockIdx.x * blockDim.x + threadIdx.x;
    if (i >= E * HEADS) return;
    int e = i >> 3, h = i & 7;
    int s = src[e], d = dst[e];
    float a = ea[e];
    const float* pl = xl + (long)s * CDIM + h * HD;
    const float* pr = xr + (long)d * CDIM + h * HD;
    const float* pw = We + h * HD;
    const float* pa = att + h * HD;
    float sc = 0.f;
#pragma unroll
    for (int j = 0; j < HD; ++j) {
        float f = pl[j] + pr[j] + a * pw[j];
        f = (f > 0.f) ? f : f * 0.2f;       // leaky_relu(0.2)
        sc += pa[j] * f;
    }
    score[i] = sc;
    atomicMax(smaxKey + (long)d * HEADS + h, f2key(sc));
}

// ---------------- edge pass 2: e = exp(score - max), segment sum -------------
__global__ __launch_bounds__(256) void k_edge_exp(float* __restrict__ score,
                                                  const unsigned* __restrict__ smaxKey,
                                                  float* __restrict__ ssum,
                                                  const int* __restrict__ dst,
                                                  int E) {
    int i = blockIdx.x * blockDim.x + threadIdx.x;
    if (i >= E * HEADS) return;
    int e = i >> 3, h = i & 7;
    int d = dst[e];
    float m = key2f(smaxKey[(long)d * HEADS + h]);
    float ev = __expf(score[i] - m);
    score[i] = ev;
    atomicAdd(ssum + (long)d * HEADS + h, ev);
}

// ---------------- edge pass 3: g[dst] += alpha * xl[src]  (one thread/element)
__global__ __launch_bounds__(256) void k_edge_aggr(const float* __restrict__ escore,
                                                   const float* __restrict__ ssum,
                                                   const float* __restrict__ xl,
                                                   const int* __restrict__ src,
                                                   const int* __restrict__ dst,
                                                   float* __restrict__ g,
                                                   int E) {
    long i = (long)blockIdx.x * blockDim.x + threadIdx.x;
    if (i >= (long)E * CDIM) return;
    int e = (int)(i >> 8);
    int c = (int)(i & 255);
    int h = c >> 5;
    int s = src[e], d = dst[e];
    float alpha = escore[(long)e * HEADS + h] /
                  (ssum[(long)d * HEADS + h] + 1e-16f);
    atomicAdd(g + (long)d * CDIM + c, alpha * xl[(long)s * CDIM + c]);
}

// ---------------- LayerNorm over 256 dims: one wave32 per row ----------------
// out = LN(in1 (+ in2) (+ biasvec)) * gam + bet
__global__ __launch_bounds__(256) void k_ln(const float* __restrict__ in1,
                                            const float* __restrict__ in2,
                                            const float* __restrict__ biasvec,
                                            const float* __restrict__ gam,
                                            const float* __restrict__ bet,
                                            float* __restrict__ out, int M) {
    int wid  = (int)((blockIdx.x * (long)blockDim.x + threadIdx.x) >> 5);
    int lane = threadIdx.x & 31;
    if (wid >= M) return;
    const long base = (long)wid * CDIM;
    float v[8];
    float s = 0.f;
#pragma unroll
    for (int j = 0; j < 8; ++j) {
        int c = lane + j * 32;
        float x = in1[base + c];
        if (in2)     x += in2[base + c];
        if (biasvec) x += biasvec[c];
        v[j] = x;
        s += x;
    }
#pragma unroll
    for (int o = 16; o > 0; o >>= 1) s += __shfl_xor(s, o, 32);
    float mu = s * (1.f / CDIM);
    float var = 0.f;
#pragma unroll
    for (int j = 0; j < 8; ++j) { float dv = v[j] - mu; var += dv * dv; }
#pragma unroll
    for (int o = 16; o > 0; o >>= 1) var += __shfl_xor(var, o, 32);
    var *= (1.f / CDIM);
    float inv = rsqrtf(var + 1e-5f);
#pragma unroll
    for (int j = 0; j < 8; ++j) {
        int c = lane + j * 32;
        out[base + c] = (v[j] - mu) * inv * gam[c] + bet[c];
    }
}

// =============================================================================
extern "C" void kernel_launch(void* const* d_in, const int* in_sizes, int n_in,
                              void* d_out, int out_size, void* d_ws, size_t ws_size,
                              hipStream_t stream) {
    const int N  = in_sizes[0] / 18;     // 20000
    const int E  = in_sizes[2];          // 320000
    const int NID = in_sizes[26];        // 1000

    const float* x_gnn = (const float*)d_in[0];
    const int*   eidx  = (const int*)d_in[1];
    const int*   src   = eidx;
    const int*   dst   = eidx + E;
    const float* eattr = (const float*)d_in[2];
    const float* W_in  = (const float*)d_in[3];
    const float* b_in  = (const float*)d_in[4];
    const float* Wl1   = (const float*)d_in[5];
    const float* Wr1   = (const float*)d_in[6];
    const float* We1   = (const float*)d_in[7];
    const float* att1  = (const float*)d_in[8];
    const float* bg1   = (const float*)d_in[9];
    const float* ln1_g = (const float*)d_in[10];
    const float* ln1_b = (const float*)d_in[11];
    const float* W_down= (const float*)d_in[12];
    const float* lnd_g = (const float*)d_in[13];
    const float* lnd_b = (const float*)d_in[14];
    const float* Wl2   = (const float*)d_in[15];
    const float* Wr2   = (const float*)d_in[16];
    const float* We2   = (const float*)d_in[17];
    const float* att2  = (const float*)d_in[18];
    const float* bg2   = (const float*)d_in[19];
    const float* ln2_g = (const float*)d_in[20];
    const float* ln2_b = (const float*)d_in[21];
    const float* W_up  = (const float*)d_in[22];
    const float* lnu_g = (const float*)d_in[23];
    const float* lnu_b = (const float*)d_in[24];
    const float* W_cls = (const float*)d_in[25];
    const float* b_cls = (const float*)d_in[26];

    float* outp   = (float*)d_out;
    float* logits = outp;                       // [N, NID]
    float* h_out  = outp + (long)N * NID;       // [N, 256]

    // ---- workspace carve-up (floats) ----
    float* w = (float*)d_ws;
    float* h0   = w;               w += (long)N * 32;
    float* xl   = w;               w += (long)N * CDIM;   // reused as zin staging
    float* xr   = w;               w += (long)N * CDIM;
    float* h1   = w;               w += (long)N * CDIM;
    float* z    = w;               w += (long)N * CDIM;
    float* g    = w;               w += (long)N * CDIM;   // aggregation / h2
    float* ssum = w;               w += (long)N * HEADS;
    unsigned* smaxKey = (unsigned*)w; w += (long)N * HEADS;
    float* esc  = w;               w += (long)E * HEADS;

    const dim3 blk(256);
    const int gemmGx = (N / 16 + 7) / 8;              // M tiles / 8 waves per block
    const dim3 gemm256(gemmGx, CDIM / 16);
    const dim3 gemmCls(gemmGx, (NID + 15) / 16);
    const int ehGrid  = (E * HEADS + 255) / 256;
    const int eaGrid  = (int)(((long)E * CDIM + 255) / 256);
    const int lnGrid  = (N + 7) / 8;

    // 1. input projection
    k_in_proj<<<(N * 32 + 255) / 256, blk, 0, stream>>>(x_gnn, W_in, b_in, h0, N);

    // 2. layer-1 source/target transforms (N x 32 @ 32 x 256)
    k_gemm_f32_wmma<<<gemm256, blk, 0, stream>>>(h0, Wl1, nullptr, xl, N, 32, CDIM);
    k_gemm_f32_wmma<<<gemm256, blk, 0, stream>>>(h0, Wr1, nullptr, xr, N, 32, CDIM);

    // 3. layer-1 attention
    hipMemsetAsync(g, 0, (size_t)N * CDIM * 4, stream);
    hipMemsetAsync(smaxKey, 0, (size_t)N * HEADS * 4, stream);  // key 0 == identity
    hipMemsetAsync(ssum, 0, (size_t)N * HEADS * 4, stream);
    k_edge_score<<<ehGrid, blk, 0, stream>>>(xl, xr, eattr, src, dst, We1, att1,
                                             esc, smaxKey, E);
    k_edge_exp<<<ehGrid, blk, 0, stream>>>(esc, smaxKey, ssum, dst, E);
    k_edge_aggr<<<eaGrid, blk, 0, stream>>>(esc, ssum, xl, src, dst, g, E);

    // 4. h1 = LN(g1 + bg1)
    k_ln<<<lnGrid, blk, 0, stream>>>(g, nullptr, bg1, ln1_g, ln1_b, h1, N);

    // 5. z = LN(h1 @ W_down)
    k_gemm_f32_wmma<<<gemm256, blk, 0, stream>>>(h1, W_down, nullptr, xl, N, CDIM, CDIM);
    k_ln<<<lnGrid, blk, 0, stream>>>(xl, nullptr, nullptr, lnd_g, lnd_b, z, N);

    // 6. layer-2 transforms (N x 256 @ 256 x 256)
    k_gemm_f32_wmma<<<gemm256, blk, 0, stream>>>(z, Wl2, nullptr, xl, N, CDIM, CDIM);
    k_gemm_f32_wmma<<<gemm256, blk, 0, stream>>>(z, Wr2, nullptr, xr, N, CDIM, CDIM);

    // 7. layer-2 attention
    hipMemsetAsync(g, 0, (size_t)N * CDIM * 4, stream);
    hipMemsetAsync(smaxKey, 0, (size_t)N * HEADS * 4, stream);
    hipMemsetAsync(ssum, 0, (size_t)N * HEADS * 4, stream);
    k_edge_score<<<ehGrid, blk, 0, stream>>>(xl, xr, eattr, src, dst, We2, att2,
                                             esc, smaxKey, E);
    k_edge_exp<<<ehGrid, blk, 0, stream>>>(esc, smaxKey, ssum, dst, E);
    k_edge_aggr<<<eaGrid, blk, 0, stream>>>(esc, ssum, xl, src, dst, g, E);

    // 8. z2 = LN(g2 + bg2 + z)   (in-place into z; each element single-owner)
    k_ln<<<lnGrid, blk, 0, stream>>>(g, z, bg2, ln2_g, ln2_b, z, N);

    // 9. h2 = z2 @ W_up  -> g
    k_gemm_f32_wmma<<<gemm256, blk, 0, stream>>>(z, W_up, nullptr, g, N, CDIM, CDIM);

    // 10. h = LN(h1 + h2)  -> second output
    k_ln<<<lnGrid, blk, 0, stream>>>(h1, g, nullptr, lnu_g, lnu_b, h_out, N);

    // 11. logits = h @ W_cls + b_cls  -> first output
    k_gemm_f32_wmma<<<gemmCls, blk, 0, stream>>>(h_out, W_cls, b_cls, logits,
                                                 N, CDIM, NID);
}